// OnlineSiameseModel_24240795419178
// MI455X (gfx1250) — compile-verified
//
#include <hip/hip_runtime.h>
#include <stdint.h>

// Problem constants (from the reference setup)
#define B_N     4096      // batch
#define FIN     16384     // 128*128 flattened image
#define D_EMB   128       // embedding dim
#define KT      64        // K tile for encode GEMM
#define NITER   (FIN / KT)
#define ALPHA   0.5f
#define NEG_INF (-1.0e30f)

typedef float        v2f __attribute__((ext_vector_type(2)));
typedef float        v8f __attribute__((ext_vector_type(8)));
typedef unsigned int v4u __attribute__((ext_vector_type(4)));
typedef int          v4i __attribute__((ext_vector_type(4)));
typedef int          v8i __attribute__((ext_vector_type(8)));

// ---------------------------------------------------------------------------
// V_WMMA_F32_16X16X4_F32 wrapper (wave32; 8-arg VOP3P form)
// A: 16x4 f32 -> 2 VGPRs/lane (lanes 0-15: M=lane, K={0,1}; lanes 16-31: M=lane-16, K={2,3})
// B: 4x16 f32 -> 2 VGPRs/lane (lanes 0-15: N=lane, K={0,1}; lanes 16-31: N=lane-16, K={2,3})
// C/D: 16x16 f32 -> 8 VGPRs (vgpr r, lanes 0-15: M=r, N=lane; lanes 16-31: M=r+8)
// ---------------------------------------------------------------------------
__device__ __forceinline__ v8f wmma_f32_16x16x4(v2f a, v2f b, v8f c) {
  return __builtin_amdgcn_wmma_f32_16x16x4_f32(false, a, false, b, (short)0, c,
                                               false, false);
}

// ---------------------------------------------------------------------------
// Tensor Data Mover: build a 2D D# (ISA ch.8) and issue TENSOR_LOAD_TO_LDS.
// 6-arg builtin (clang-23 / therock-10.0 flavor):
//   (uint32x4 g0, int32x8 g1, int32x4 g2, int32x4 g3, int32x8 pad, i32 cpol)
// Group2/3 zero (2D tensor). Optional LDS padding to avoid bank conflicts.
// ---------------------------------------------------------------------------
__device__ __forceinline__ void tdm_load_2d_to_lds(
    uint32_t lds_off, const void* gptr,
    uint32_t tensor_d0, uint32_t tensor_d1, uint64_t stride0_elems,
    uint32_t tile_d0, uint32_t tile_d1,
    uint32_t pad_en, uint32_t pad_interval_code, uint32_t pad_amount_code)
{
  const uint64_t ga = (uint64_t)(uintptr_t)gptr;
  v4u g0;
  g0[0] = 1u;                                     // count=1 (valid user desc)
  g0[1] = lds_off;                                // lds_addr (bytes)
  g0[2] = (uint32_t)ga;                           // global_addr[31:0]
  g0[3] = (uint32_t)((ga >> 32) & 0x01FFFFFFull)  // global_addr[56:32]
        | (2u << 30);                             // type=2 ("image")
  v8i g1;
  uint32_t w0 = (2u << 16)                        // data_size = 4 bytes
              | ((pad_en & 1u) << 20)
              | ((pad_interval_code & 7u) << 22)
              | ((pad_amount_code & 0x7Fu) << 25);
  g1[0] = (int)w0;                                                  // mask=0 (no cluster)
  g1[1] = (int)((tensor_d0 & 0xFFFFu) << 16);                       // dim0[15:0] @ 63:48
  g1[2] = (int)(((tensor_d0 >> 16) & 0xFFFFu) | ((tensor_d1 & 0xFFFFu) << 16));
  g1[3] = (int)(((tensor_d1 >> 16) & 0xFFFFu) | ((tile_d0 & 0xFFFFu) << 16));
  g1[4] = (int)(tile_d1 & 0xFFFFu);                                 // tile_dim2 = 0
  g1[5] = (int)(uint32_t)(stride0_elems & 0xFFFFFFFFull);           // dim0_stride[31:0]
  g1[6] = (int)(uint32_t)((stride0_elems >> 32) & 0xFFFFull);       // dim0_stride[47:32]
  g1[7] = 0;
  v4i gz4 = {0, 0, 0, 0};
  v8i gz8 = {0, 0, 0, 0, 0, 0, 0, 0};
  __builtin_amdgcn_tensor_load_to_lds(g0, g1, gz4, gz4, gz8, 0);
}

// ---------------------------------------------------------------------------
// Stateless hash -> uniform(0,1) -> Gumbel noise
// ---------------------------------------------------------------------------
__device__ __forceinline__ uint32_t hash3(uint32_t a, uint32_t b, uint32_t c) {
  uint32_t h = a * 0x9E3779B1u + b * 0x85EBCA77u + c * 0xC2B2AE3Du + 0x27D4EB2Fu;
  h ^= h >> 16; h *= 0x7FEB352Du;
  h ^= h >> 15; h *= 0x846CA68Bu;
  h ^= h >> 16;
  return h;
}
__device__ __forceinline__ float gumbel_from(uint32_t h) {
  float u = ((float)h + 0.5f) * 2.3283064365386963e-10f;  // (0,1)
  return -__logf(-__logf(u));
}

// ---------------------------------------------------------------------------
// Kernel 1: embs = X @ W + b   (memory-bound: 256 MB of X, ~11.5 us floor)
// grid = 256 blocks (one 16-row M strip each), block = 256 threads = 8 waves.
// TDM ping-pong double buffering: issue tile t+1 into buf[(t+1)&1], then
// s_wait_tensorcnt 2 (in-order TENSORcnt => tile t's two DMAs retired while
// tile t+1's two stay in flight), barrier, compute tile t. DMA overlaps WMMA.
// X tile [16 x 64] padded to LDS stride 65; W tile [64 x 128] padded to 130.
// Wave w computes N columns [16w, 16w+16) with v_wmma_f32_16x16x4_f32.
// ---------------------------------------------------------------------------
#define XS_STRIDE 65
#define WS_STRIDE 130
#define XS_ELEMS  (16 * XS_STRIDE)
#define WS_ELEMS  (KT * WS_STRIDE)

__global__ void __launch_bounds__(256)
encode_gemm_kernel(const float* __restrict__ X, const float* __restrict__ W,
                   const float* __restrict__ bias, float* __restrict__ embs)
{
  __shared__ float Xs[2][XS_ELEMS];
  __shared__ float Ws[2][WS_ELEMS];

  const int tid  = threadIdx.x;
  const int wid  = tid >> 5;
  const int lane = tid & 31;
  const int half = lane >> 4;
  const int l15  = lane & 15;
  const int m0   = blockIdx.x * 16;

  const uint32_t xs_off0 = (uint32_t)(uintptr_t)(&Xs[0][0]);
  const uint32_t xs_off1 = (uint32_t)(uintptr_t)(&Xs[1][0]);
  const uint32_t ws_off0 = (uint32_t)(uintptr_t)(&Ws[0][0]);
  const uint32_t ws_off1 = (uint32_t)(uintptr_t)(&Ws[1][0]);

  v8f acc = {0.f, 0.f, 0.f, 0.f, 0.f, 0.f, 0.f, 0.f};
  const int col = wid * 16 + l15;

  // Prologue: stage tile 0 into buffer 0
  if (wid == 0) {
    tdm_load_2d_to_lds(xs_off0, X + (size_t)m0 * FIN,
                       FIN, B_N, (uint64_t)FIN, KT, 16, 1u, 5u, 0u);
    tdm_load_2d_to_lds(ws_off0, W,
                       D_EMB, FIN, (uint64_t)D_EMB, D_EMB, KT, 1u, 6u, 1u);
  }

  for (int it = 0; it < NITER; ++it) {
    const int buf = it & 1;
    if (wid == 0) {
      if (it + 1 < NITER) {
        // Prefetch tile it+1 into the other buffer (its previous readers
        // were released by the trailing barrier of iteration it-1).
        const int k1 = (it + 1) * KT;
        tdm_load_2d_to_lds(buf ? xs_off0 : xs_off1,
                           X + (size_t)m0 * FIN + k1,
                           FIN, B_N, (uint64_t)FIN, KT, 16, 1u, 5u, 0u);
        tdm_load_2d_to_lds(buf ? ws_off0 : ws_off1,
                           W + (size_t)k1 * D_EMB,
                           D_EMB, FIN, (uint64_t)D_EMB, D_EMB, KT, 1u, 6u, 1u);
        __builtin_amdgcn_s_wait_tensorcnt(2);  // tile it retired; it+1 in flight
      } else {
        __builtin_amdgcn_s_wait_tensorcnt(0);  // last tile: drain
      }
    }
    __syncthreads();  // tile it visible to all 8 waves

    const float* __restrict__ Xsb = &Xs[buf][0];
    const float* __restrict__ Wsb = &Ws[buf][0];
#pragma unroll
    for (int kk = 0; kk < KT; kk += 4) {
      const int ka = kk + 2 * half;
      v2f a, b;
      a[0] = Xsb[l15 * XS_STRIDE + ka];
      a[1] = Xsb[l15 * XS_STRIDE + ka + 1];
      b[0] = Wsb[ka * WS_STRIDE + col];
      b[1] = Wsb[(ka + 1) * WS_STRIDE + col];
      acc = wmma_f32_16x16x4(a, b, acc);
    }
    __syncthreads();  // all reads of buf done before it is re-issued at it+2
  }

  const float bv = bias[col];
#pragma unroll
  for (int r = 0; r < 8; ++r) {
    const int m = m0 + r + 8 * half;
    embs[(size_t)m * D_EMB + col] = acc[r] + bv;
  }
}

// ---------------------------------------------------------------------------
// Kernel 2: sq[i] = ||embs[i]||^2   (one wave32 per row)
// ---------------------------------------------------------------------------
__global__ void __launch_bounds__(256)
rowsq_kernel(const float* __restrict__ embs, float* __restrict__ sq)
{
  const int wid = threadIdx.x >> 5, lane = threadIdx.x & 31;
  const int row = blockIdx.x * 8 + wid;
  const float4 v = ((const float4*)(embs + (size_t)row * D_EMB))[lane];
  float s = v.x * v.x + v.y * v.y + v.z * v.z + v.w * v.w;
#pragma unroll
  for (int m = 16; m >= 1; m >>= 1) s += __shfl_xor(s, m, 32);
  if (lane == 0) sq[row] = s;
}

// ---------------------------------------------------------------------------
// Kernel 3: positive mining. pos = gumbel-argmax over same-label j (incl. self),
// then d_pos[i] = sq[i] + sq[p] - 2*dot(embs[i], embs[p]). One wave per row.
// ---------------------------------------------------------------------------
__global__ void __launch_bounds__(256)
pos_mine_kernel(const float* __restrict__ embs, const float* __restrict__ sq,
                const int* __restrict__ labels, float* __restrict__ d_pos)
{
  const int wid = threadIdx.x >> 5, lane = threadIdx.x & 31;
  const int i = blockIdx.x * 8 + wid;
  const int li = labels[i];

  float bg = NEG_INF; int bj = i;
  for (int j = lane; j < B_N; j += 32) {
    if (labels[j] == li) {
      const float g = gumbel_from(hash3(0x9E3779B9u, (uint32_t)i, (uint32_t)j));
      if (g > bg) { bg = g; bj = j; }
    }
  }
#pragma unroll
  for (int m = 16; m >= 1; m >>= 1) {
    const float og = __shfl_xor(bg, m, 32);
    const int   oj = __shfl_xor(bj, m, 32);
    if (og > bg) { bg = og; bj = oj; }
  }
  const int p = __shfl(bj, 0, 32);

  const float4 a = ((const float4*)(embs + (size_t)i * D_EMB))[lane];
  const float4 c = ((const float4*)(embs + (size_t)p * D_EMB))[lane];
  float dt = a.x * c.x + a.y * c.y + a.z * c.z + a.w * c.w;
#pragma unroll
  for (int m = 16; m >= 1; m >>= 1) dt += __shfl_xor(dt, m, 32);
  if (lane == 0) d_pos[i] = sq[i] + sq[p] - 2.0f * dt;
}

// ---------------------------------------------------------------------------
// Kernel 4: negative mining + loss, WMMA Gram tiles (Dmat never materialized).
// Block = 16 anchor rows, 8 waves; wave w scans j-stripe [jt*128+16w, +16).
// Per 16x16 tile: 32x v_wmma_f32_16x16x4_f32 (K=128), then semi-hard /
// random gumbel running-argmax per row held in registers; shfl_xor reduce
// within half-waves (masks 1..8 keep wave32 halves independent), LDS combine.
// ---------------------------------------------------------------------------
__global__ void __launch_bounds__(256)
neg_mine_kernel(const float* __restrict__ embs, const float* __restrict__ sq,
                const float* __restrict__ dpos, const int* __restrict__ labels,
                float* __restrict__ accum)
{
  __shared__ float s_sqi[16];
  __shared__ float s_dp[16];
  __shared__ int   s_lab[16];
  __shared__ float s_sh_g[16][8];
  __shared__ float s_sh_D[16][8];
  __shared__ float s_rn_g[16][8];
  __shared__ float s_rn_D[16][8];

  const int tid = threadIdx.x;
  const int wid = tid >> 5, lane = tid & 31, half = lane >> 4, l15 = lane & 15;
  const int i0 = blockIdx.x * 16;

  if (tid < 16) {
    s_sqi[tid] = sq[i0 + tid];
    s_dp[tid]  = dpos[i0 + tid];
    s_lab[tid] = labels[i0 + tid];
  }
  __syncthreads();

  // A fragments: 16 anchor rows x K=128 (stays in 64 VGPRs, embs is L2-hot)
  v2f afrag[32];
  const float* arow = embs + (size_t)(i0 + l15) * D_EMB + 2 * half;
#pragma unroll
  for (int kk = 0; kk < D_EMB; kk += 4) {
    afrag[kk >> 2][0] = arow[kk];
    afrag[kk >> 2][1] = arow[kk + 1];
  }

  float bsh_g[8], bsh_D[8], brn_g[8], brn_D[8];
#pragma unroll
  for (int r = 0; r < 8; ++r) {
    bsh_g[r] = NEG_INF; bsh_D[r] = 0.f;
    brn_g[r] = NEG_INF; brn_D[r] = 0.f;
  }

  for (int jt = 0; jt < B_N / 128; ++jt) {
    const int j = jt * 128 + wid * 16 + l15;
    const float* brow = embs + (size_t)j * D_EMB + 2 * half;
    v8f acc = {0.f, 0.f, 0.f, 0.f, 0.f, 0.f, 0.f, 0.f};
#pragma unroll
    for (int kk = 0; kk < D_EMB; kk += 4) {
      v2f bb;
      bb[0] = brow[kk];
      bb[1] = brow[kk + 1];
      acc = wmma_f32_16x16x4(afrag[kk >> 2], bb, acc);
    }
    const float sqj  = sq[j];
    const int   labj = labels[j];
#pragma unroll
    for (int r = 0; r < 8; ++r) {
      const int   irow = r + 8 * half;
      const float Dij  = s_sqi[irow] + sqj - 2.0f * acc[r];
      const float dp   = s_dp[irow];
      if (labj != s_lab[irow]) {
        const uint32_t iu = (uint32_t)(i0 + irow), ju = (uint32_t)j;
        const float grn = gumbel_from(hash3(0x85EBCA6Bu, iu, ju));
        if (grn > brn_g[r]) { brn_g[r] = grn; brn_D[r] = Dij; }
        if (dp < Dij && Dij < dp + ALPHA) {
          const float gsh = gumbel_from(hash3(0xC2B2AE35u, iu, ju));
          if (gsh > bsh_g[r]) { bsh_g[r] = gsh; bsh_D[r] = Dij; }
        }
      }
    }
  }

#pragma unroll
  for (int r = 0; r < 8; ++r) {
#pragma unroll
    for (int m = 1; m < 16; m <<= 1) {
      float og = __shfl_xor(bsh_g[r], m, 32);
      float oD = __shfl_xor(bsh_D[r], m, 32);
      if (og > bsh_g[r]) { bsh_g[r] = og; bsh_D[r] = oD; }
      og = __shfl_xor(brn_g[r], m, 32);
      oD = __shfl_xor(brn_D[r], m, 32);
      if (og > brn_g[r]) { brn_g[r] = og; brn_D[r] = oD; }
    }
    if (l15 == 0) {
      const int irow = r + 8 * half;
      s_sh_g[irow][wid] = bsh_g[r]; s_sh_D[irow][wid] = bsh_D[r];
      s_rn_g[irow][wid] = brn_g[r]; s_rn_D[irow][wid] = brn_D[r];
    }
  }
  __syncthreads();

  if (tid < 16) {
    float bg = NEG_INF, bD = 0.f, rg = NEG_INF, rD = 0.f;
#pragma unroll
    for (int w = 0; w < 8; ++w) {
      if (s_sh_g[tid][w] > bg) { bg = s_sh_g[tid][w]; bD = s_sh_D[tid][w]; }
      if (s_rn_g[tid][w] > rg) { rg = s_rn_g[tid][w]; rD = s_rn_D[tid][w]; }
    }
    const float d_an = (bg > 0.5f * NEG_INF) ? bD : rD;  // has_semihard ? sh : rand
    const float loss = s_dp[tid] - d_an + ALPHA;
    if (loss > 0.0f) atomicAdd(accum, loss);
  }
}

// ---------------------------------------------------------------------------
__global__ void zero_kernel(float* accum) { *accum = 0.0f; }
__global__ void finalize_kernel(const float* __restrict__ accum, float* __restrict__ out) {
  out[0] = accum[0] * (1.0f / (float)B_N);
}

// ---------------------------------------------------------------------------
extern "C" void kernel_launch(void* const* d_in, const int* in_sizes, int n_in,
                              void* d_out, int out_size, void* d_ws, size_t ws_size,
                              hipStream_t stream)
{
  const float* X      = (const float*)d_in[0];  // [4096,1,128,128] f32
  const float* W      = (const float*)d_in[1];  // [16384,128] f32
  const float* bias   = (const float*)d_in[2];  // [128] f32
  const int*   labels = (const int*)d_in[3];    // [4096] i32

  char*  ws    = (char*)d_ws;
  float* embs  = (float*)ws;                              // 2 MB
  float* sq    = (float*)(ws + (size_t)B_N * D_EMB * 4);  // 16 KB
  float* dpos  = sq + B_N;                                // 16 KB
  float* accum = dpos + B_N;                              // 4 B

  zero_kernel<<<1, 1, 0, stream>>>(accum);
  encode_gemm_kernel<<<B_N / 16, 256, 0, stream>>>(X, W, bias, embs);
  rowsq_kernel<<<B_N / 8, 256, 0, stream>>>(embs, sq);
  pos_mine_kernel<<<B_N / 8, 256, 0, stream>>>(embs, sq, labels, dpos);
  neg_mine_kernel<<<B_N / 16, 256, 0, stream>>>(embs, sq, dpos, labels, accum);
  finalize_kernel<<<1, 1, 0, stream>>>(accum, (float*)d_out);
}